// SparseTransFusionHead1_44349832299023
// MI455X (gfx1250) — compile-verified
//
#include <hip/hip_runtime.h>

typedef float v2f __attribute__((ext_vector_type(2)));
typedef float v4f __attribute__((ext_vector_type(4)));
typedef float v8f __attribute__((ext_vector_type(8)));
typedef int   v4i __attribute__((ext_vector_type(4)));
typedef unsigned long long u64;
typedef unsigned int u32;

static constexpr int cB = 4, cN = 30000, cC = 128, cNC = 10, cKV = 5000, cP = 200;
static constexpr int NKEY = cN * cNC;        // 300000 keys per batch
static constexpr int CAND = 8192;            // candidate/sort capacity (power of 2)
static constexpr float BN_S = 0.9999950000374997f;   // 1/sqrt(1+1e-5)

#if defined(__has_builtin)
#if __has_builtin(__builtin_amdgcn_global_load_async_to_lds_b128) && \
    __has_builtin(__builtin_amdgcn_s_wait_asynccnt)
#define HAVE_ASYNC_LDS 1
#endif
#endif
#ifndef HAVE_ASYNC_LDS
#define HAVE_ASYNC_LDS 0
#endif

static __device__ __forceinline__ v8f wmma4(v2f a, v2f b, v8f c) {
  // D(16x16,f32) = A(16x4,f32) x B(4x16,f32) + C
  return __builtin_amdgcn_wmma_f32_16x16x4_f32(false, a, false, b, (short)0, c, false, false);
}

#if HAVE_ASYNC_LDS
static __device__ __forceinline__ void async_g2l_b128(const float* g, float* l) {
  __builtin_amdgcn_global_load_async_to_lds_b128(
      (__attribute__((address_space(1))) v4i*)(g),
      (__attribute__((address_space(3))) v4i*)(l), 0, 0);
}
#endif

// ---------------------------------------------------------------------------
// 1) Heatmap MLP (128->128 BN+ReLU -> 128->10 sigmoid), emits sort keys + hist
//    w1 staged pair-interleaved so each B fragment is one ds_load_b64.
// ---------------------------------------------------------------------------
__global__ __launch_bounds__(256) void k_heatmap(
    const float* __restrict__ feats, const float* __restrict__ w1,
    const float* __restrict__ b1, const float* __restrict__ g1,
    const float* __restrict__ be1, const float* __restrict__ w2,
    const float* __restrict__ b2, u64* __restrict__ keys, u32* __restrict__ hist) {
  extern __shared__ char dynsm[];
  float* w1P = (float*)dynsm;           // [32 grp][128 col][2 hi][2 e] = 16384
  float* w2s = w1P + 128 * 128;         // [10][128]
  float* Xt  = w2s + 10 * 128;          // [16][132]
  float* Ht  = Xt + 16 * 132;           // [16][129]
  const int tid = threadIdx.x;
  for (int i = tid; i < 128 * 128; i += 256) {
    const int n = i >> 7, k = i & 127;
    const int g = k >> 2, j = k & 3;
    w1P[(g * 128 + n) * 4 + (j & 1) * 2 + (j >> 1)] = w1[i];
  }
  for (int i = tid; i < 10 * 128; i += 256) w2s[i] = w2[i];
  __syncthreads();
  const int w = tid >> 5, lane = tid & 31, hi = lane >> 4, lo = lane & 15;
  const int col = w * 16 + lo;
  const int ntile = cN / 16;                       // 1875 per batch
  for (int t = blockIdx.x; t < cB * ntile; t += gridDim.x) {
    const int b = t / ntile;
    const int r0 = (t % ntile) * 16;
#if HAVE_ASYNC_LDS
    for (int i = tid; i < 512; i += 256) {         // 16 rows x 32 float4
      const int m = i >> 5, k4 = (i & 31) * 4;
      async_g2l_b128(feats + (size_t)(b * cN + r0 + m) * cC + k4, &Xt[m * 132 + k4]);
    }
    __builtin_amdgcn_s_wait_asynccnt(0);
#else
    for (int i = tid; i < 16 * 128; i += 256) {
      const int m = i >> 7, k = i & 127;
      Xt[m * 132 + k] = feats[(size_t)(b * cN + r0 + m) * cC + k];
    }
#endif
    // prefetch next tile (emits global_prefetch_b8)
    if (tid < 16 && (t + gridDim.x) < cB * ntile) {
      const int t2 = t + gridDim.x;
      const int b2 = t2 / ntile, r2 = (t2 % ntile) * 16;
      __builtin_prefetch(&feats[(size_t)(b2 * cN + r2 + tid) * cC], 0, 1);
    }
    __syncthreads();
    v8f acc; for (int r = 0; r < 8; ++r) acc[r] = 0.f;
    const float* xr = &Xt[lo * 132 + 2 * hi];
    const float* wpb = &w1P[col * 4 + hi * 2];
#pragma unroll
    for (int ks = 0; ks < 32; ++ks) {
      const v2f a = *(const v2f*)(xr + ks * 4);
      const v2f bb = *(const v2f*)(wpb + ks * 512);
      acc = wmma4(a, bb, acc);
    }
    const float gj = g1[col], bj = be1[col], b1j = b1[col];
    for (int r = 0; r < 8; ++r) {
      const int m = r + 8 * hi;
      const float y = acc[r] + b1j;
      Ht[m * 129 + col] = fmaxf(gj * (y * BN_S) + bj, 0.f);
    }
    __syncthreads();
    if (tid < 160) {
      const int row = tid / 10, cls = tid % 10;
      float z = b2[cls];
      const float* hr = &Ht[row * 129];
      const float* wr = &w2s[cls * 128];
      for (int k = 0; k < 128; ++k) z += hr[k] * wr[k];
      const float heat = 1.f / (1.f + expf(-z));
      const u32 bits = __float_as_uint(heat);      // positive float -> monotone bits
      const int flat = (r0 + row) * cNC + cls;
      keys[(size_t)b * NKEY + flat] = ((u64)bits << 32) | (u32)(~(u32)flat);
      atomicAdd(&hist[b * 65536 + (bits >> 16)], 1u);
    }
    __syncthreads();
  }
}

// ---------------------------------------------------------------------------
// 2) Histogram suffix scan -> cutoff bucket for top-KV
// ---------------------------------------------------------------------------
__global__ void k_scan(const u32* __restrict__ hist, int* __restrict__ cut) {
  const int b = blockIdx.x, tid = threadIdx.x;       // 256 threads
  __shared__ u32 psum[256];
  const u32* hb = hist + b * 65536;
  u32 s = 0;
  for (int i = 0; i < 256; ++i) s += hb[tid * 256 + i];
  psum[tid] = s;
  __syncthreads();
  if (tid == 0) {
    u32 cum = 0; int gsel = 0;
    for (int g = 255; g >= 0; --g) {
      if (cum + psum[g] >= (u32)cKV) { gsel = g; break; }
      cum += psum[g];
      if (g == 0) gsel = 0;
    }
    int cu = gsel * 256; u32 above = cum;
    for (int bin = gsel * 256 + 255; bin >= gsel * 256; --bin) {
      u32 h = hb[bin];
      if (cum + h >= (u32)cKV) { cu = bin; above = cum; break; }
      cum += h; cu = bin; above = cum;
    }
    cut[b * 2 + 0] = cu; cut[b * 2 + 1] = (int)above;
  }
}

__global__ void k_compact(const u64* __restrict__ keys, const int* __restrict__ cut,
                          u32* __restrict__ cnt, u64* __restrict__ cand) {
  const int t = blockIdx.x * blockDim.x + threadIdx.x;
  if (t >= cB * NKEY) return;
  const int b = t / NKEY, i = t % NKEY;
  const u64 key = keys[(size_t)b * NKEY + i];
  if ((int)(key >> 48) >= cut[b * 2]) {
    const u32 pos = atomicAdd(&cnt[b], 1u);
    if (pos < CAND) cand[(size_t)b * CAND + pos] = key;
  }
}

// ---------------------------------------------------------------------------
// 3) In-LDS bitonic sort (descending) of 8192 u64 keys per batch
// ---------------------------------------------------------------------------
__global__ void k_sort(u64* __restrict__ buf) {
  extern __shared__ char dynsm[];
  u64* s = (u64*)dynsm;                              // 8192 * 8B = 64KB
  u64* bb = buf + (size_t)blockIdx.x * CAND;
  const int tid = threadIdx.x, nt = blockDim.x;
  for (int i = tid; i < CAND; i += nt) s[i] = bb[i];
  __syncthreads();
  for (unsigned k = 2; k <= CAND; k <<= 1) {
    for (unsigned j = k >> 1; j > 0; j >>= 1) {
      for (int i = tid; i < CAND; i += nt) {
        const unsigned ixj = i ^ j;
        if (ixj > (unsigned)i) {
          const u64 a = s[i], b2 = s[ixj];
          const bool up = ((i & k) == 0);
          if (up ? (a < b2) : (a > b2)) { s[i] = b2; s[ixj] = a; }   // descending
        }
      }
      __syncthreads();
    }
  }
  for (int i = tid; i < CAND; i += nt) bb[i] = s[i];
}

__global__ void k_extract_kv(const u64* __restrict__ cand, int* __restrict__ kvidx) {
  const int t = blockIdx.x * blockDim.x + threadIdx.x;
  if (t >= cB * cKV) return;
  const int b = t / cKV, i = t % cKV;
  kvidx[t] = (int)(~(u32)cand[(size_t)b * CAND + i]);
}

__global__ void k_propbuild(const int* __restrict__ kvidx, u64* __restrict__ prop) {
  const int t = blockIdx.x * blockDim.x + threadIdx.x;
  if (t >= cB * CAND) return;
  const int b = t / CAND, i = t % CAND;
  prop[t] = (i < cKV) ? ((((u64)(u32)kvidx[b * cKV + i]) << 32) | (u32)i) : 0ull;
}

__global__ void k_extract_prop(const u64* __restrict__ prop, int* __restrict__ topprop) {
  const int t = blockIdx.x * blockDim.x + threadIdx.x;
  if (t >= cB * cP) return;
  const int b = t / cP, p = t % cP;
  topprop[t] = (int)(u32)(prop[(size_t)b * CAND + p] & 0xffffffffull);
}

// ---------------------------------------------------------------------------
// 4) Gathers
// ---------------------------------------------------------------------------
__global__ void k_gather(const float* __restrict__ feats, const int* __restrict__ sidx,
                         const int* __restrict__ kvidx, float* __restrict__ kvfeat,
                         float* __restrict__ kvpos) {
  const int bi = blockIdx.x;                          // [0, B*KV)
  const int b = bi / cKV;
  const int c = threadIdx.x;
  const int row = kvidx[bi] / cNC;
  kvfeat[(size_t)bi * cC + c] = feats[(size_t)(b * cN + row) * cC + c];
  if (c < 2) kvpos[bi * 2 + c] = (float)sidx[(size_t)(b * cN + row) * 2 + c] * 0.5f + 0.5f;
}

__global__ void k_qprep(const int* __restrict__ topprop, const float* __restrict__ kvfeat,
                        const float* __restrict__ kvpos, const float* __restrict__ ce_w,
                        const float* __restrict__ ce_b, float* __restrict__ qfeat,
                        float* __restrict__ qpos) {
  const int bp = blockIdx.x;                          // [0, B*P)
  const int b = bp / cP;
  const int c = threadIdx.x;
  const int tp = topprop[bp];
  const int rk = tp / cNC, cls = tp % cNC;
  qfeat[(size_t)bp * cC + c] = kvfeat[(size_t)(b * cKV + rk) * cC + c] + ce_w[c * cNC + cls] + ce_b[c];
  if (c < 2) qpos[bp * 2 + c] = kvpos[(b * cKV + rk) * 2 + (1 - c)];   // flip xy
}

// ---------------------------------------------------------------------------
// 5) Position-embed layer1 (Conv1d(2,C,1) + BN + ReLU)
// ---------------------------------------------------------------------------
__global__ void k_pos1(const float* __restrict__ pos, int Lrow, int rev,
                       const float* __restrict__ w1, const float* __restrict__ b1,
                       const float* __restrict__ g, const float* __restrict__ be,
                       float* __restrict__ H, int rows) {
  const int t = blockIdx.x * blockDim.x + threadIdx.x;
  if (t >= rows * cC) return;
  const int r = t >> 7, c = t & 127;
  const int b = r / Lrow, rr = r % Lrow;
  const int pr = rev ? (b * Lrow + (Lrow - 1 - rr)) : r;
  const float p0 = pos[pr * 2 + 0], p1 = pos[pr * 2 + 1];
  const float y = p0 * w1[c * 2 + 0] + p1 * w1[c * 2 + 1] + b1[c];
  H[(size_t)r * cC + c] = fmaxf(g[c] * (y * BN_S) + be[c], 0.f);
}

// ---------------------------------------------------------------------------
// 6) Generic f32 WMMA GEMM:  Y[:, yc0:yc0+OC] = ep( (X[+X2]) @ W^T + bias )
//    block = 256 thr (8 waves x 16 rows), grid.y = OC/16 col tiles
//    ep: 0=none, 1=relu, 2=relu(g*((v+bias)*BN_S)+be)
// ---------------------------------------------------------------------------
__global__ __launch_bounds__(256) void k_gemm(
    const float* __restrict__ X, const float* __restrict__ X2,
    const float* __restrict__ W, const float* __restrict__ bias,
    const float* __restrict__ g, const float* __restrict__ be,
    float* __restrict__ Y, int M, int K, int ldx, int ldy, int yc0, int ep) {
  __shared__ float WtP[64 * 64];                     // [K/4 grp][16 n][2 hi][2 e], K<=256
  __shared__ float Xt[128 * 36];                     // [m][k32] padded, 16B-aligned rows
  const int tid = threadIdx.x;
  const int r0 = blockIdx.x * 128, c0 = blockIdx.y * 16;
  for (int i = tid; i < 16 * K; i += 256) {
    const int n = i / K, k = i % K;
    const int gg = k >> 2, j = k & 3;
    WtP[gg * 64 + n * 4 + (j & 1) * 2 + (j >> 1)] = W[(size_t)(c0 + n) * K + k];
  }
  __syncthreads();
  const int w = tid >> 5, lane = tid & 31, hi = lane >> 4, lo = lane & 15;
  v8f acc; for (int r = 0; r < 8; ++r) acc[r] = 0.f;
  for (int kk = 0; kk < K; kk += 32) {
#if HAVE_ASYNC_LDS
    if (X2 == nullptr) {
      for (int i = tid; i < 128 * 8; i += 256) {     // 128 rows x 8 float4
        const int m = i >> 3, k4 = (i & 7) * 4;
        int row = r0 + m; if (row >= M) row = M - 1;
        async_g2l_b128(X + (size_t)row * ldx + kk + k4, &Xt[m * 36 + k4]);
      }
      __builtin_amdgcn_s_wait_asynccnt(0);
    } else
#endif
    {
      for (int i = tid; i < 128 * 32; i += 256) {
        const int m = i >> 5, k = i & 31;
        int row = r0 + m; if (row >= M) row = M - 1;
        float v = X[(size_t)row * ldx + kk + k];
        if (X2) v += X2[(size_t)row * ldx + kk + k];
        Xt[m * 36 + k] = v;
      }
    }
    __syncthreads();
    const float* xr = &Xt[(w * 16 + lo) * 36 + 2 * hi];
    const float* wpb = &WtP[(kk >> 2) * 64 + lo * 4 + hi * 2];
#pragma unroll
    for (int ks = 0; ks < 8; ++ks) {
      const v2f a = *(const v2f*)(xr + ks * 4);
      const v2f bb = *(const v2f*)(wpb + ks * 64);
      acc = wmma4(a, bb, acc);
    }
    __syncthreads();
  }
  const float bi = bias ? bias[c0 + lo] : 0.f;
  const float gg = (ep == 2) ? g[c0 + lo] : 0.f;
  const float bb2 = (ep == 2) ? be[c0 + lo] : 0.f;
  const int col = yc0 + c0 + lo;
  for (int r = 0; r < 8; ++r) {
    const int row = r0 + w * 16 + r + 8 * hi;
    if (row < M) {
      float v = acc[r] + bi;
      if (ep == 1) v = fmaxf(v, 0.f);
      else if (ep == 2) v = fmaxf(gg * (v * BN_S) + bb2, 0.f);
      Y[(size_t)row * ldy + col] = v;
    }
  }
}

// ---------------------------------------------------------------------------
// 7) Flash-style attention, hd=16, NH=8 (one wave per head), f32 WMMA
// ---------------------------------------------------------------------------
__global__ __launch_bounds__(256) void k_attn(
    const float* __restrict__ Q, int ldq, const float* __restrict__ Kp, int ldk,
    const float* __restrict__ Vp, int ldv, float* __restrict__ O, int ldo,
    int Lq, int Lk) {
  const int nqt = (Lq + 15) / 16;
  const int b = blockIdx.x / nqt, qt = blockIdx.x % nqt;
  const int tid = threadIdx.x, h = tid >> 5, lane = tid & 31, hi = lane >> 4, lo = lane & 15;
  __shared__ float Pt[8][16][18];                    // pad 18 -> b64-aligned reads
  const int i0 = qt * 16;
  int qrow = i0 + lo; if (qrow >= Lq) qrow = Lq - 1;
  const float* qp = Q + (size_t)(b * Lq + qrow) * ldq + h * 16;
  v2f qa[4];
#pragma unroll
  for (int f = 0; f < 4; ++f) {                      // fold 1/sqrt(16) into Q
    const v2f qv = *(const v2f*)(qp + f * 4 + 2 * hi);
    qa[f].x = qv.x * 0.25f; qa[f].y = qv.y * 0.25f;
  }
  v8f oacc; for (int r = 0; r < 8; ++r) oacc[r] = 0.f;
  float mR[8], lR[8];
  for (int r = 0; r < 8; ++r) { mR[r] = -3.0e38f; lR[r] = 0.f; }
  const int nkt = (Lk + 15) / 16;
  for (int kt = 0; kt < nkt; ++kt) {
    const int kj0 = kt * 16;
    int krow = kj0 + lo; if (krow >= Lk) krow = Lk - 1;
    const float* kr = Kp + (size_t)(b * Lk + krow) * ldk + h * 16;
    v8f S; for (int r = 0; r < 8; ++r) S[r] = 0.f;
#pragma unroll
    for (int f = 0; f < 4; ++f) {
      const v4f kv = *(const v4f*)(kr + f * 4);
      v2f bb; bb.x = hi ? kv.y : kv.x; bb.y = hi ? kv.w : kv.z;
      S = wmma4(qa[f], bb, S);
    }
    if (kj0 + lo >= Lk) for (int r = 0; r < 8; ++r) S[r] = -3.0e38f;  // mask cols
#pragma unroll
    for (int r = 0; r < 8; ++r) {
      float v = S[r];
      for (int msk = 1; msk < 16; msk <<= 1) v = fmaxf(v, __shfl_xor(v, msk, 32));
      const float mn = fmaxf(mR[r], v);
      const float al = __expf(mR[r] - mn);
      const float p = __expf(S[r] - mn);
      float ps = p;
      for (int msk = 1; msk < 16; msk <<= 1) ps += __shfl_xor(ps, msk, 32);
      lR[r] = lR[r] * al + ps;
      mR[r] = mn;
      oacc[r] *= al;
      Pt[h][r + 8 * hi][lo] = p;                     // wave-private LDS slice
    }
#pragma unroll
    for (int f = 0; f < 4; ++f) {
      const int k0 = f * 4;
      const v2f a = *(const v2f*)&Pt[h][lo][k0 + 2 * hi];
      int vr0 = kj0 + k0 + hi;     if (vr0 >= Lk) vr0 = Lk - 1;
      int vr1 = kj0 + k0 + 2 + hi; if (vr1 >= Lk) vr1 = Lk - 1;
      v2f bb;
      bb.x = Vp[(size_t)(b * Lk + vr0) * ldv + h * 16 + lo];
      bb.y = Vp[(size_t)(b * Lk + vr1) * ldv + h * 16 + lo];
      oacc = wmma4(a, bb, oacc);
    }
  }
  for (int r = 0; r < 8; ++r) {
    const int row = i0 + r + 8 * hi;
    if (row < Lq) O[(size_t)(b * Lq + row) * ldo + h * 16 + lo] = oacc[r] / lR[r];
  }
}

// ---------------------------------------------------------------------------
// 8) LayerNorm with residual: Y = LN(A + R) * g + b   (C = 128, 1 row/block)
// ---------------------------------------------------------------------------
__global__ void k_ln(const float* __restrict__ A, const float* __restrict__ R,
                     const float* __restrict__ g, const float* __restrict__ bta,
                     float* __restrict__ Y) {
  const int row = blockIdx.x, c = threadIdx.x;
  __shared__ float red1[4], red2[4];
  const float v = A[(size_t)row * cC + c] + R[(size_t)row * cC + c];
  float s = v;
  for (int msk = 1; msk < 32; msk <<= 1) s += __shfl_xor(s, msk, 32);
  if ((c & 31) == 0) red1[c >> 5] = s;
  __syncthreads();
  const float mean = (red1[0] + red1[1] + red1[2] + red1[3]) * (1.f / 128.f);
  const float d = v - mean;
  float q = d * d;
  for (int msk = 1; msk < 32; msk <<= 1) q += __shfl_xor(q, msk, 32);
  if ((c & 31) == 0) red2[c >> 5] = q;
  __syncthreads();
  const float var = (red2[0] + red2[1] + red2[2] + red2[3]) * (1.f / 128.f);
  Y[(size_t)row * cC + c] = d * rsqrtf(var + 1e-5f) * g[c] + bta[c];
}

// ---------------------------------------------------------------------------
// 9) Final head projection 64->20 channels, +query_pos, [B,P,20] -> [B,20,P]
// ---------------------------------------------------------------------------
__global__ void k_final(const float* __restrict__ hh, const float* __restrict__ w2,
                        const float* __restrict__ b2, const float* __restrict__ qpos,
                        float* __restrict__ out) {
  const int bp = blockIdx.x;                          // [0, B*P)
  const int b = bp / cP, p = bp % cP;
  const int t = threadIdx.x;
  if (t >= 20) return;
  const int HOF[20] = {0,0,1,2,2,2,3,3,4,4,5,5,5,5,5,5,5,5,5,5};
  const int OOF[20] = {0,1,0,0,1,2,0,1,0,1,0,1,2,3,4,5,6,7,8,9};
  const int hd = HOF[t], oc = OOF[t];
  const float* hrow = hh + (size_t)(hd * cB * cP + bp) * 64;
  const float* wr = w2 + (size_t)(hd * 10 + oc) * 64;
  float v = b2[hd * 10 + oc];
  for (int k = 0; k < 64; ++k) v += hrow[k] * wr[k];
  if (t < 2) v += qpos[bp * 2 + t];
  out[(size_t)b * 20 * cP + t * cP + p] = v;
}

// ---------------------------------------------------------------------------
// Host orchestration
// ---------------------------------------------------------------------------
extern "C" void kernel_launch(void* const* d_in, const int* in_sizes, int n_in,
                              void* d_out, int out_size, void* d_ws, size_t ws_size,
                              hipStream_t stream) {
  (void)in_sizes; (void)n_in; (void)out_size; (void)ws_size;
  int a = 0;
  const float* feats   = (const float*)d_in[a++];
  const int*   sidx    = (const int*)  d_in[a++];
  const float* hm_w1   = (const float*)d_in[a++];
  const float* hm_b1   = (const float*)d_in[a++];
  const float* hm_g1   = (const float*)d_in[a++];
  const float* hm_be1  = (const float*)d_in[a++];
  const float* hm_w2   = (const float*)d_in[a++];
  const float* hm_b2   = (const float*)d_in[a++];
  const float* ce_w    = (const float*)d_in[a++];
  const float* ce_b    = (const float*)d_in[a++];
  const float* pes_w1  = (const float*)d_in[a++];
  const float* pes_b1  = (const float*)d_in[a++];
  const float* pes_g   = (const float*)d_in[a++];
  const float* pes_be  = (const float*)d_in[a++];
  const float* pes_w2  = (const float*)d_in[a++];
  const float* pes_b2  = (const float*)d_in[a++];
  const float* pec_w1  = (const float*)d_in[a++];
  const float* pec_b1  = (const float*)d_in[a++];
  const float* pec_g   = (const float*)d_in[a++];
  const float* pec_be  = (const float*)d_in[a++];
  const float* pec_w2  = (const float*)d_in[a++];
  const float* pec_b2  = (const float*)d_in[a++];
  const float* sa_wqkv = (const float*)d_in[a++];
  const float* sa_bqkv = (const float*)d_in[a++];
  const float* sa_wo   = (const float*)d_in[a++];
  const float* sa_bo   = (const float*)d_in[a++];
  const float* ca_wqkv = (const float*)d_in[a++];
  const float* ca_bqkv = (const float*)d_in[a++];
  const float* ca_wo   = (const float*)d_in[a++];
  const float* ca_bo   = (const float*)d_in[a++];
  const float* ln1_g   = (const float*)d_in[a++];
  const float* ln1_b   = (const float*)d_in[a++];
  const float* ln2_g   = (const float*)d_in[a++];
  const float* ln2_b   = (const float*)d_in[a++];
  const float* ln3_g   = (const float*)d_in[a++];
  const float* ln3_b   = (const float*)d_in[a++];
  const float* ffn_w1  = (const float*)d_in[a++];
  const float* ffn_b1  = (const float*)d_in[a++];
  const float* ffn_w2  = (const float*)d_in[a++];
  const float* ffn_b2  = (const float*)d_in[a++];
  const float* head_w1 = (const float*)d_in[a++];
  const float* head_g  = (const float*)d_in[a++];
  const float* head_be = (const float*)d_in[a++];
  const float* head_w2 = (const float*)d_in[a++];
  const float* head_b2 = (const float*)d_in[a++];
  float* out = (float*)d_out;

  char* wp = (char*)d_ws;
  size_t off = 0;
  auto alloc = [&](size_t bytes) -> char* {
    off = (off + 255) & ~(size_t)255;
    char* p = wp + off; off += bytes; return p;
  };
  u64*   keys    = (u64*)  alloc((size_t)cB * NKEY * 8);
  u32*   hist    = (u32*)  alloc((size_t)cB * 65536 * 4);
  int*   cut     = (int*)  alloc(cB * 2 * 4);
  u32*   cnt     = (u32*)  alloc(cB * 4);
  u64*   cand    = (u64*)  alloc((size_t)cB * CAND * 8);
  u64*   prop    = (u64*)  alloc((size_t)cB * CAND * 8);
  int*   kvidx   = (int*)  alloc((size_t)cB * cKV * 4);
  int*   topprop = (int*)  alloc(cB * cP * 4);
  float* kvfeat  = (float*)alloc((size_t)cB * cKV * cC * 4);
  float* kvpos   = (float*)alloc((size_t)cB * cKV * 2 * 4);
  float* qfeat   = (float*)alloc((size_t)cB * cP * cC * 4);
  float* qpos    = (float*)alloc((size_t)cB * cP * 2 * 4);
  float* qpe_h   = (float*)alloc((size_t)cB * cP * cC * 4);
  float* qpe     = (float*)alloc((size_t)cB * cP * cC * 4);
  float* kpe_h   = (float*)alloc((size_t)cB * cKV * cC * 4);
  float* kpe     = (float*)alloc((size_t)cB * cKV * cC * 4);
  float* sqkv    = (float*)alloc((size_t)cB * cP * 384 * 4);
  float* abuf    = (float*)alloc((size_t)cB * cP * cC * 4);
  float* t1      = (float*)alloc((size_t)cB * cP * cC * 4);
  float* q1      = (float*)alloc((size_t)cB * cP * cC * 4);
  float* cq      = (float*)alloc((size_t)cB * cP * cC * 4);
  float* ck      = (float*)alloc((size_t)cB * cKV * cC * 4);
  float* cv      = (float*)alloc((size_t)cB * cKV * cC * 4);
  float* q2      = (float*)alloc((size_t)cB * cP * cC * 4);
  float* f1      = (float*)alloc((size_t)cB * cP * 256 * 4);
  float* f2      = (float*)alloc((size_t)cB * cP * cC * 4);
  float* q3      = (float*)alloc((size_t)cB * cP * cC * 4);
  float* hh      = (float*)alloc((size_t)6 * cB * cP * 64 * 4);

  auto gemm = [&](const float* X, const float* X2, const float* W, const float* bias,
                  const float* g, const float* be, float* Y,
                  int M, int K, int ldx, int ldy, int yc0, int OC, int ep) {
    dim3 grid((M + 127) / 128, OC / 16);
    k_gemm<<<grid, 256, 0, stream>>>(X, X2, W, bias, g, be, Y, M, K, ldx, ldy, yc0, ep);
  };

  // --- top-K selection prep ---
  (void)hipMemsetAsync(hist, 0, (size_t)cB * 65536 * 4, stream);
  (void)hipMemsetAsync(cnt, 0, cB * 4, stream);
  (void)hipMemsetAsync(cand, 0, (size_t)cB * CAND * 8, stream);

  const size_t hm_lds = (size_t)(128 * 128 + 10 * 128 + 16 * 132 + 16 * 129) * 4;
  k_heatmap<<<240, 256, hm_lds, stream>>>(feats, hm_w1, hm_b1, hm_g1, hm_be1,
                                          hm_w2, hm_b2, keys, hist);
  k_scan<<<cB, 256, 0, stream>>>(hist, cut);
  k_compact<<<(cB * NKEY + 255) / 256, 256, 0, stream>>>(keys, cut, cnt, cand);
  k_sort<<<cB, 512, CAND * 8, stream>>>(cand);
  k_extract_kv<<<(cB * cKV + 255) / 256, 256, 0, stream>>>(cand, kvidx);
  k_gather<<<cB * cKV, 128, 0, stream>>>(feats, sidx, kvidx, kvfeat, kvpos);
  k_propbuild<<<(cB * CAND + 255) / 256, 256, 0, stream>>>(kvidx, prop);
  k_sort<<<cB, 512, CAND * 8, stream>>>(prop);
  k_extract_prop<<<(cB * cP + 255) / 256, 256, 0, stream>>>(prop, topprop);
  k_qprep<<<cB * cP, 128, 0, stream>>>(topprop, kvfeat, kvpos, ce_w, ce_b, qfeat, qpos);

  // --- position embeddings ---
  const int MQ = cB * cP, MK = cB * cKV;
  k_pos1<<<(MQ * cC + 255) / 256, 256, 0, stream>>>(qpos, cP, 0, pes_w1, pes_b1, pes_g, pes_be, qpe_h, MQ);
  gemm(qpe_h, nullptr, pes_w2, pes_b2, nullptr, nullptr, qpe, MQ, 128, 128, 128, 0, 128, 0);
  k_pos1<<<(MK * cC + 255) / 256, 256, 0, stream>>>(kvpos, cKV, 1, pec_w1, pec_b1, pec_g, pec_be, kpe_h, MK);
  gemm(kpe_h, nullptr, pec_w2, pec_b2, nullptr, nullptr, kpe, MK, 128, 128, 128, 0, 128, 0);

  // --- self-attention ---
  gemm(qfeat, qpe, sa_wqkv, sa_bqkv, nullptr, nullptr, sqkv, MQ, 128, 128, 384, 0, 256, 0);       // q,k from q+qpe
  gemm(qfeat, nullptr, sa_wqkv + 256 * 128, sa_bqkv + 256, nullptr, nullptr, sqkv, MQ, 128, 128, 384, 256, 128, 0); // v from q
  k_attn<<<cB * ((cP + 15) / 16), 256, 0, stream>>>(sqkv, 384, sqkv + 128, 384, sqkv + 256, 384, abuf, 128, cP, cP);
  gemm(abuf, nullptr, sa_wo, sa_bo, nullptr, nullptr, t1, MQ, 128, 128, 128, 0, 128, 0);
  k_ln<<<MQ, 128, 0, stream>>>(qfeat, t1, ln1_g, ln1_b, q1);

  // --- cross-attention ---
  gemm(q1, qpe, ca_wqkv, ca_bqkv, nullptr, nullptr, cq, MQ, 128, 128, 128, 0, 128, 0);
  gemm(kvfeat, kpe, ca_wqkv + 128 * 128, ca_bqkv + 128, nullptr, nullptr, ck, MK, 128, 128, 128, 0, 128, 0);
  gemm(kvfeat, nullptr, ca_wqkv + 256 * 128, ca_bqkv + 256, nullptr, nullptr, cv, MK, 128, 128, 128, 0, 128, 0);
  k_attn<<<cB * ((cP + 15) / 16), 256, 0, stream>>>(cq, 128, ck, 128, cv, 128, abuf, 128, cP, cKV);
  gemm(abuf, nullptr, ca_wo, ca_bo, nullptr, nullptr, t1, MQ, 128, 128, 128, 0, 128, 0);
  k_ln<<<MQ, 128, 0, stream>>>(q1, t1, ln2_g, ln2_b, q2);

  // --- FFN ---
  gemm(q2, nullptr, ffn_w1, ffn_b1, nullptr, nullptr, f1, MQ, 128, 128, 256, 0, 256, 1);
  gemm(f1, nullptr, ffn_w2, ffn_b2, nullptr, nullptr, f2, MQ, 256, 256, 128, 0, 128, 0);
  k_ln<<<MQ, 128, 0, stream>>>(q2, f2, ln3_g, ln3_b, q3);

  // --- heads ---
  for (int i = 0; i < 6; ++i) {
    gemm(q3, nullptr, head_w1 + (size_t)i * 64 * 128, nullptr,
         head_g + i * 64, head_be + i * 64, hh + (size_t)i * MQ * 64,
         MQ, 128, 128, 64, 0, 64, 2);
  }
  k_final<<<cB * cP, 32, 0, stream>>>(hh, head_w2, head_b2, qpos, out);
}